// CPCUnsupersivedCriterion_35115652612346
// MI455X (gfx1250) — compile-verified
//
#include <hip/hip_runtime.h>
#include <hip/hip_bf16.h>
#include <stdint.h>

// CPC unsupervised criterion for MI455X (gfx1250), wave32 + V_WMMA_F32_16X16X4_F32.
//
// Shapes (fixed by the reference): B=32, S=128, dAR=dEnc=256, K=12, N=128, W=116.
// Phase 1: locC[k][b][w][e] = sum_a cFeature[b][w][a] * Wp[k][e][a]   (WMMA GEMM)
// Phase 2: per (b,w): scores = (locC rows over k) @ (gathered neg rows)^T via WMMA,
//          fused pos-dot, logsumexp + argmax, atomic reduce into 24 output floats.

typedef __attribute__((ext_vector_type(2))) float v2f;
typedef __attribute__((ext_vector_type(8))) float v8f;

#define CB   32      // batch
#define CS   128     // sequence
#define DAR  256     // dAR
#define DE   256     // dEnc
#define CK   12      // K predictions
#define CN   128     // N negatives
#define CW   116     // W = S - K
#define WT   8       // ceil(W/16) w-tiles per batch row
#define WPAD 128     // padded w extent in workspace (WT*16)

__device__ __forceinline__ v8f wmma_f32_4(v2f a, v2f b, v8f c) {
  // D = A(16x4 f32) * B(4x16 f32) + C(16x16 f32)
  return __builtin_amdgcn_wmma_f32_16x16x4_f32(false, a, false, b, (short)0, c,
                                               false, false);
}

// ---------------------------------------------------------------------------
// Phase 0: zero the 24 accumulated outputs (losses[12] ++ acc[12]).
// ---------------------------------------------------------------------------
__global__ void cpc_zero(float* __restrict__ out) {
  if (threadIdx.x < 2 * CK) out[threadIdx.x] = 0.0f;
}

// ---------------------------------------------------------------------------
// Phase 1: locC GEMM.  grid = K*B*WT blocks, 128 threads (4 waves) per block.
// Each block: fixed (k, b, wtile).  A-tile (16 w-rows x 256 a) staged in LDS,
// each wave computes a 16(w) x 64(e) output tile: 64 K-steps x 4 WMMAs.
// ---------------------------------------------------------------------------
__global__ __launch_bounds__(128) void cpc_locc(const float* __restrict__ cF,
                                                const float* __restrict__ Wp,
                                                float* __restrict__ locC) {
  __shared__ float ldsA[16 * DAR];  // 16 KB

  const int blk = blockIdx.x;
  const int wt  = blk & 7;
  const int b   = (blk >> 3) & 31;
  const int k   = blk >> 8;
  const int tid = threadIdx.x;

  // Cooperative stage of the A tile (cFeature rows w0..w0+15, all 256 a's).
  {
    const int row = tid >> 3;            // 16 rows, 8 threads per row
    const int col = (tid & 7) * 32;      // 32 consecutive floats per thread
    const float4* src =
        (const float4*)(cF + ((size_t)(b * CS + wt * 16 + row) * DAR + col));
    float4* dst = (float4*)&ldsA[row * DAR + col];
#pragma unroll
    for (int i = 0; i < 8; ++i) dst[i] = src[i];
  }
  __syncthreads();

  const int lane  = tid & 31;
  const int wave  = tid >> 5;
  const int ebase = wave * 64;           // each wave owns 64 e-columns
  const int hi    = lane >> 4;           // half-wave selects K-pair {0,1}/{2,3}
  const int ln    = lane & 15;           // A-row (w) / B-col (e) within tile

  v8f acc[4];
#pragma unroll
  for (int j = 0; j < 4; ++j) acc[j] = v8f{0.f, 0.f, 0.f, 0.f, 0.f, 0.f, 0.f, 0.f};

  const float* wpk = Wp + (size_t)k * DE * DAR;
  const float* aP  = &ldsA[ln * DAR + 2 * hi];

  for (int a0 = 0; a0 < DAR; a0 += 4) {
    v2f af = *(const v2f*)(aP + a0);  // A frag: lanes hold c[w0+ln][a0+2*hi .. +1]
#pragma unroll
    for (int j = 0; j < 4; ++j) {
      // B frag: B[a][n] = Wp[k][ebase + j*16 + n][a]
      v2f bf = *(const v2f*)&wpk[(size_t)(ebase + j * 16 + ln) * DAR + a0 + 2 * hi];
      acc[j] = wmma_f32_4(af, bf, acc[j]);
    }
  }

  // Scatter 16x64 result tile to workspace: locC[k][b][wp][e]
  float* dstBase = locC + (((size_t)k * CB + b) * WPAD + wt * 16) * DE;
#pragma unroll
  for (int j = 0; j < 4; ++j) {
#pragma unroll
    for (int r = 0; r < 8; ++r) {
      const int m = r + 8 * hi;  // w within tile
      dstBase[(size_t)m * DE + (ebase + j * 16 + ln)] = acc[j][r];
    }
  }
}

// ---------------------------------------------------------------------------
// Phase 2: per-(b,w) scores + logsumexp.  grid = B*W blocks, 1 wave each.
// A = locC rows over k (M=16, rows >= 12 zeroed), B = gathered negative rows.
// ---------------------------------------------------------------------------
__global__ __launch_bounds__(32) void cpc_score(const float* __restrict__ enc,
                                                const float* __restrict__ locC,
                                                const long long* __restrict__ extIdx,
                                                float* __restrict__ out) {
  __shared__ float sneg[16][CN];  // 8 KB: scores[k][n]

  const int blk  = blockIdx.x;
  const int w    = blk % CW;
  const int b    = blk / CW;
  const int lane = threadIdx.x;
  const int hi   = lane >> 4;
  const int ln   = lane & 15;

  // Preload gather byte-offsets: n-tile nt, column ln -> row extIdx[b][nt*16+ln][w]
  int idxOff[8];
#pragma unroll
  for (int nt = 0; nt < 8; ++nt) {
    long long idx =
        extIdx[(size_t)b * CN * CW + (size_t)(nt * 16 + ln) * CW + w];
    idxOff[nt] = (int)idx * (DE * 4);
  }

  const bool kvalid = (ln < CK);
  const float* arow = locC + (((size_t)ln * CB + b) * WPAD + w) * DE;
  const float* prow = enc + (size_t)(b * CS + w + ln + 1) * DE;  // pos row for k=ln
  const char* encB  = (const char*)enc;

  v8f acc[8];
#pragma unroll
  for (int nt = 0; nt < 8; ++nt)
    acc[nt] = v8f{0.f, 0.f, 0.f, 0.f, 0.f, 0.f, 0.f, 0.f};
  float posAcc = 0.0f;

  for (int e0 = 0; e0 < DE; e0 += 4) {
    v2f af = v2f{0.f, 0.f};
    if (kvalid) {
      af = *(const v2f*)&arow[e0 + 2 * hi];
      // fused positive dot: lane (ln, hi) covers k=ln, e = e0+2*hi, e0+2*hi+1
      v2f p = *(const v2f*)&prow[e0 + 2 * hi];
      posAcc += af.x * p.x + af.y * p.y;
    }
#pragma unroll
    for (int nt = 0; nt < 8; ++nt) {
      v2f bf = *(const v2f*)(encB + idxOff[nt] + (size_t)(e0 + 2 * hi) * 4);
      acc[nt] = wmma_f32_4(af, bf, acc[nt]);
    }
  }

  const float inv_e = 1.0f / (float)DE;
#pragma unroll
  for (int nt = 0; nt < 8; ++nt) {
#pragma unroll
    for (int r = 0; r < 8; ++r) {
      sneg[r + 8 * hi][nt * 16 + ln] = acc[nt][r] * inv_e;  // row k, col n
    }
  }
  __syncthreads();

  // Combine the two e-halves of the positive dot; lane k ends up with posScore[k].
  posAcc += __shfl_xor(posAcc, 16);
  posAcc *= inv_e;

  const float invBW = 1.0f / (float)(CB * CW);
  for (int k = 0; k < CK; ++k) {
    const float pk = __shfl(posAcc, k);
    const float4 nv = *(const float4*)&sneg[k][lane * 4];  // 4 negs per lane
    float mn = fmaxf(fmaxf(nv.x, nv.y), fmaxf(nv.z, nv.w));
#pragma unroll
    for (int off = 16; off >= 1; off >>= 1) mn = fmaxf(mn, __shfl_xor(mn, off));
    const float m = fmaxf(mn, pk);
    float s = __expf(nv.x - m) + __expf(nv.y - m) + __expf(nv.z - m) +
              __expf(nv.w - m);
#pragma unroll
    for (int off = 16; off >= 1; off >>= 1) s += __shfl_xor(s, off);
    s += __expf(pk - m);
    if (lane == 0) {
      const float lse = m + __logf(s);
      atomicAdd(&out[k], (lse - pk) * invBW);
      atomicAdd(&out[CK + k], (pk >= mn ? 1.0f : 0.0f) * invBW);
    }
  }
}

// ---------------------------------------------------------------------------
extern "C" void kernel_launch(void* const* d_in, const int* in_sizes, int n_in,
                              void* d_out, int out_size, void* d_ws, size_t ws_size,
                              hipStream_t stream) {
  (void)in_sizes; (void)n_in; (void)out_size; (void)ws_size;
  const float* cF        = (const float*)d_in[0];       // (B,S,dAR) f32
  const float* enc       = (const float*)d_in[1];       // (B,S,dEnc) f32
  const float* Wp        = (const float*)d_in[2];       // (K,dEnc,dAR) f32
  const long long* extIdx = (const long long*)d_in[3];  // (N*W*B,) i64
  float* out  = (float*)d_out;                          // 24 floats
  float* locC = (float*)d_ws;                           // 12*32*128*256 f32 = 48 MB

  cpc_zero<<<1, 32, 0, stream>>>(out);
  cpc_locc<<<CK * CB * WT, 128, 0, stream>>>(cF, Wp, locC);
  cpc_score<<<CB * CW, 32, 0, stream>>>(enc, locC, extIdx, out);
}